// TwoTower_37615323578740
// MI455X (gfx1250) — compile-verified
//
#include <hip/hip_runtime.h>

// CDNA5 / gfx1250, wave32.
typedef __attribute__((ext_vector_type(2))) float v2f;
typedef __attribute__((ext_vector_type(8))) float v8f;

#define EMBED_DIM 64
#define TILE 16          // rows (dot products) per wave
#define WAVES_PER_BLOCK 8

// One wave32 computes 16 row-wise dot products <xe[r,:], ye[r,:]> for a tile of
// 16 batch rows using V_WMMA_F32_16X16X4_F32 accumulation over K=64 (16 chunks
// of K=4). Only the diagonal of the 16x16 result is used; the wasted FLOPs are
// free because the kernel is HBM-gather bound (~8.6 MB @ 23.3 TB/s).
__global__ __launch_bounds__(WAVES_PER_BLOCK * 32)
void twotower_wmma_kernel(const int* __restrict__ x,
                          const int* __restrict__ y,
                          const float* __restrict__ Wc,
                          const float* __restrict__ Wa,
                          float* __restrict__ out,
                          int batch)
{
    const int lane = threadIdx.x & 31;
    const int wave = threadIdx.x >> 5;
    const int tile = blockIdx.x * WAVES_PER_BLOCK + wave;
    const int row0 = tile * TILE;
    if (row0 >= batch) return;   // whole-wave exit; EXEC stays all-1s for WMMA

    // A (16x4 f32): lane l holds M = l&15, K = k0 + 2*(l>>4) + {0,1} in 2 VGPRs.
    // B (4x16 f32): symmetric striping -> lane l holds N = l&15, same K pair.
    // => identical per-lane addressing for both matrices: one float2 load each.
    const int m   = lane & 15;           // A row / B column within the tile
    const int khi = (lane >> 4) << 1;    // +0 for lanes 0-15, +2 for lanes 16-31

    int r = row0 + m;
    if (r >= batch) r = batch - 1;       // safety clamp (batch % 16 == 0 normally)

    const long long xi = x[r];
    const long long yi = y[r];
    const float* __restrict__ arow = Wc + xi * (long long)EMBED_DIM;
    const float* __restrict__ brow = Wa + yi * (long long)EMBED_DIM;

    // Issue all 32 b64 gathers up front; lanes l and l+16 jointly cover each
    // 256-B embedding row exactly once across the 16 K-chunks.
    v2f a[16], b[16];
#pragma unroll
    for (int j = 0; j < 16; ++j) {
        const int k = j * 4 + khi;
        a[j] = *(const v2f*)(arow + k);
        b[j] = *(const v2f*)(brow + k);
    }

    // Accumulate D = sum_j A_j * B_j in exact fp32 via WMMA 16x16x4.
    v8f c = {};
#pragma unroll
    for (int j = 0; j < 16; ++j) {
        // 8 args: (neg_a, A, neg_b, B, c_mod, C, reuse_a, reuse_b)
        c = __builtin_amdgcn_wmma_f32_16x16x4_f32(
                false, a[j], false, b[j], (short)0, c, false, false);
    }

    // C/D 16x16 f32 layout: VGPR v, lanes 0-15 -> (M=v,   N=lane),
    //                               lanes 16-31 -> (M=v+8, N=lane-16).
    // diag[i], i<8 : VGPR i,   lane i
    // diag[i], i>=8: VGPR i-8, lane i+16
    if (lane < 8) {
        out[row0 + lane] = c[lane];
    } else if (lane >= 24) {
        out[row0 + (lane - 16)] = c[lane - 24];
    }
}

extern "C" void kernel_launch(void* const* d_in, const int* in_sizes, int n_in,
                              void* d_out, int out_size, void* d_ws, size_t ws_size,
                              hipStream_t stream) {
    // setup_inputs() order: x (int32), y (int32), W_cust (f32), W_art (f32)
    const int*   x  = (const int*)d_in[0];
    const int*   y  = (const int*)d_in[1];
    const float* Wc = (const float*)d_in[2];
    const float* Wa = (const float*)d_in[3];
    float* out = (float*)d_out;

    const int batch  = in_sizes[0];                 // 16384
    const int tiles  = (batch + TILE - 1) / TILE;   // 1024
    const int blocks = (tiles + WAVES_PER_BLOCK - 1) / WAVES_PER_BLOCK;  // 128

    twotower_wmma_kernel<<<blocks, WAVES_PER_BLOCK * 32, 0, stream>>>(
        x, y, Wc, Wa, out, batch);
}